// BundleAdjustmentModel_1632087572560
// MI455X (gfx1250) — compile-verified
//
#include <hip/hip_runtime.h>
#include <hip/hip_bf16.h>
#include <math.h>

typedef __attribute__((ext_vector_type(2))) float v2f;
typedef __attribute__((ext_vector_type(8))) float v8f;

#define MIN_FOCAL    50.0f
#define MIN_DISTANCE 0.25f
#define Z_EPS        1.0e-4f
#define NGROUPS      16          // 64 views / 4 views-per-WMMA

__device__ __forceinline__ float softplusf(float x) {
    return x > 20.0f ? x : log1pf(expf(x));
}

// ---------------------------------------------------------------------------
// Setup: build WMMA A-operands in the exact per-lane register layout of
// V_WMMA_F32_16X16X4_F32. Rows carry [R | t] (homogeneous K=3 column), with
// row 0 pre-scaled by -focal and row 1 by +focal so the hot kernel's
// projection is just safe_z -> rcp -> fma(+cx/+cy).
// ws layout: group g: lane L owns {A0,A1} at ws[g*64 + 2L + {0,1}]
// A layout (32-bit A 16x4): VGPR0: lane m = A[m][0], lane m+16 = A[m][2]
//                           VGPR1: lane m = A[m][1], lane m+16 = A[m][3]
// Row packing per group g: rows 0-2 -> view 4g+0, rows 3-5 -> view 4g+1,
//                          rows 8-10 -> view 4g+2, rows 11-13 -> view 4g+3.
// ---------------------------------------------------------------------------
__global__ void ba_setup_kernel(const float* __restrict__ euler,
                                const float* __restrict__ txy,
                                const float* __restrict__ tdepth,
                                const float* __restrict__ focal_raw,
                                float* __restrict__ ws) {
    const int t = threadIdx.x;
    if (t >= NGROUPS * 32) return;

    const int g    = t >> 5;
    const int lane = t & 31;
    const int row  = lane & 15;
    const bool hi  = lane >= 16;

    int q = -1, i = 0;
    if (row < 6)                    { q = row / 3;           i = row % 3; }
    else if (row >= 8 && row < 14)  { q = 2 + (row - 8) / 3; i = (row - 8) % 3; }

    float a0 = 0.0f, a1 = 0.0f;
    if (q >= 0) {
        const int v = g * 4 + q;
        const float ex = euler[v*3+0], ey = euler[v*3+1], ez = euler[v*3+2];
        const float cxa = cosf(ex), sxa = sinf(ex);
        const float cya = cosf(ey), sya = sinf(ey);
        const float cza = cosf(ez), sza = sinf(ez);
        // M1 = Ry * Rz
        const float m00 =  cya*cza, m01 = -cya*sza, m02 = sya;
        const float m10 =  sza,     m11 =  cza,     m12 = 0.0f;
        const float m20 = -sya*cza, m21 =  sya*sza, m22 = cya;
        // R = Rx * M1 ; augment with translation column
        float R[3][4];
        R[0][0] = m00;                 R[0][1] = m01;                 R[0][2] = m02;
        R[1][0] = cxa*m10 - sxa*m20;   R[1][1] = cxa*m11 - sxa*m21;   R[1][2] = cxa*m12 - sxa*m22;
        R[2][0] = sxa*m10 + cxa*m20;   R[2][1] = sxa*m11 + cxa*m21;   R[2][2] = sxa*m12 + cxa*m22;
        R[0][3] = txy[v*2+0];
        R[1][3] = txy[v*2+1];
        R[2][3] = -(softplusf(tdepth[v]) + MIN_DISTANCE);
        // Fold projection scale into rows 0 (u: -f) and 1 (v: +f)
        const float f = softplusf(focal_raw[0]) + MIN_FOCAL;
        const float rowScale = (i == 0) ? -f : (i == 1) ? f : 1.0f;
        const int k0 = hi ? 2 : 0;
        const int k1 = hi ? 3 : 1;
        a0 = R[i][k0] * rowScale;
        a1 = R[i][k1] * rowScale;
    }
    ws[g*64 + lane*2 + 0] = a0;
    ws[g*64 + lane*2 + 1] = a1;
}

// ---------------------------------------------------------------------------
// Main: one wave per 16 points. B = 16 homogeneous points (lo half lanes:
// x,y ; hi half lanes: z,1). 16 WMMAs sweep all 64 views. Each lane reads two
// complete (fu*x, fv*y, z) triples from its own d[0..5] (zero cross-lane
// traffic), does safe_z -> v_rcp_f32 -> 2x fma, and issues two non-temporal
// 8B stores via a 32-bit running byte offset (saddr addressing, no 64-bit
// adds, no EXEC branching: out-of-range lanes are clamped to point N-1 and
// redundantly store identical values).
// ---------------------------------------------------------------------------
__global__ void __launch_bounds__(256)
ba_project_kernel(const float* __restrict__ points,
                  const float* __restrict__ ws,
                  const int* __restrict__ cxp,
                  const int* __restrict__ cyp,
                  float* __restrict__ out,
                  int N) {
    const int lane   = threadIdx.x & 31;
    const int wave   = blockIdx.x * (blockDim.x >> 5) + (threadIdx.x >> 5);
    const int nwaves = (N + 15) >> 4;
    if (wave >= nwaves) return;                 // wave-uniform: EXEC stays full

    const int col = lane & 15;
    int n = (wave << 4) + col;
    if (n >= N) n = N - 1;                      // clamp (N%16==0 -> no-op)
    const bool hi = lane >= 16;

    // Build B (4x16 homogeneous points), 2 VGPRs
    const float px = points[3*n + 0];
    const float py = points[3*n + 1];
    const float pz = points[3*n + 2];
    v2f b;
    b.x = hi ? pz   : px;    // K=2 : K=0
    b.y = hi ? 1.0f : py;    // K=3 : K=1

    // Preload all 16 A operands (32 VGPRs)
    v2f A[NGROUPS];
#pragma unroll
    for (int g = 0; g < NGROUPS; ++g)
        A[g] = *(const v2f*)(ws + g*64 + lane*2);

    const float cxf = (float)cxp[0];
    const float cyf = (float)cyp[0];

    // 32-bit running byte offsets (max 63*N*8 + n*8 < 2^31)
    const unsigned vstride = (unsigned)N * 8u;            // bytes per view
    unsigned boff = (hi ? 2u * vstride : 0u) + (unsigned)n * 8u;

    const v8f c = {0.f, 0.f, 0.f, 0.f, 0.f, 0.f, 0.f, 0.f};
    char* outb = (char*)out;

#pragma unroll
    for (int g = 0; g < NGROUPS; ++g) {
        v8f d = __builtin_amdgcn_wmma_f32_16x16x4_f32(
                    false, A[g], false, b, (short)0, c, false, false);
        // view (4g + hi*2 + 0): (fu*x, fv*y, z) = (d0, d1, d2)
        {
            const float z   = d[2];
            const float sz  = copysignf(fmaxf(fabsf(z), Z_EPS), z);
            const float inv = __builtin_amdgcn_rcpf(sz);
            v2f r;
            r.x = fmaf(d[0], inv, cxf);
            r.y = fmaf(d[1], inv, cyf);
            __builtin_nontemporal_store(r, (v2f*)(outb + boff));
        }
        // view (4g + hi*2 + 1): (d3, d4, d5)
        {
            const float z   = d[5];
            const float sz  = copysignf(fmaxf(fabsf(z), Z_EPS), z);
            const float inv = __builtin_amdgcn_rcpf(sz);
            v2f r;
            r.x = fmaf(d[3], inv, cxf);
            r.y = fmaf(d[4], inv, cyf);
            __builtin_nontemporal_store(r, (v2f*)(outb + boff + vstride));
        }
        boff += vstride * 4u;                              // advance 4 views
    }
}

extern "C" void kernel_launch(void* const* d_in, const int* in_sizes, int n_in,
                              void* d_out, int out_size, void* d_ws, size_t ws_size,
                              hipStream_t stream) {
    const float* points    = (const float*)d_in[0];   // (N,3)
    const float* euler     = (const float*)d_in[1];   // (V,3)
    const float* txy       = (const float*)d_in[2];   // (V,2)
    const float* tdepth    = (const float*)d_in[3];   // (V,)
    const float* focal_raw = (const float*)d_in[4];   // (1,)
    const int*   cxp       = (const int*)d_in[5];
    const int*   cyp       = (const int*)d_in[6];
    float*       out       = (float*)d_out;
    float*       ws        = (float*)d_ws;

    const int N = in_sizes[0] / 3;

    ba_setup_kernel<<<1, 512, 0, stream>>>(euler, txy, tdepth, focal_raw, ws);

    const int nwaves        = (N + 15) >> 4;
    const int wavesPerBlock = 8;                      // 256 threads = 8 waves
    const int blocks        = (nwaves + wavesPerBlock - 1) / wavesPerBlock;
    ba_project_kernel<<<blocks, 256, 0, stream>>>(points, ws, cxp, cyp, out, N);
}